// RieBatchNet_89593017794975
// MI455X (gfx1250) — compile-verified
//
#include <hip/hip_runtime.h>
#include <hip/hip_bf16.h>
#include <math.h>

// ---------------------------------------------------------------------------
// SPDNet forward on MI455X (gfx1250, wave32, WMMA).
// All matrix functions (sqrtm/isqrtm/logm/expm/ReEig) are computed with
// matmul-only Newton-Schulz / scaling-squaring iterations so the entire
// workload maps onto V_WMMA_F32_16X16X4_F32 (pure-f32 matrix pipe).
// Dims padded to 64; pad block carries identity (fixed point of NS iters).
// One sample per 256-thread workgroup, matrices LDS-resident (dynamic LDS,
// WGP supports 320KB/workgroup per CDNA5 ISA).
// Each wave owns a 32x16 macro-tile so both WMMAs per k-step share the
// B fragment, minimizing LDS issue pressure against the WMMA pipe.
// ---------------------------------------------------------------------------

typedef __attribute__((ext_vector_type(2))) float v2f;
typedef __attribute__((ext_vector_type(8))) float v8f;

// Only check on the device pass: amdgcn builtins are not registered on host.
#if defined(__HIP_DEVICE_COMPILE__)
#if !__has_builtin(__builtin_amdgcn_wmma_f32_16x16x4_f32)
#error "gfx1250 device pass: f32 WMMA builtin missing"
#endif
#endif

#define WG    256
#define LDSS  68            // LDS row stride in floats (padded, 8B aligned)
#define MS    (LDSS * 64)   // 4352 floats per LDS matrix buffer
#define NMAT  4096          // 64x64 global matrix footprint (floats)
#define NSAMP 8192

// ---------------------------------------------------------------------------
// Wave-cooperative 64x64x64 f32 GEMM: D = A * op(B), operands in LDS.
// 8 waves/WG; wave w computes rows [tm0, tm0+32) x cols [tn, tn+16):
//   tn = (w&3)*16, tm0 = (w>>2)*32.  Two accumulators share each B fragment.
// A-frag layout (32-bit A 16x4): lanes 0-15 K={k0,k0+1}, lanes 16-31 K={k0+2,k0+3}.
// B-frag mirrored; C/D per documented 16x16 f32 layout.
// Caller must __syncthreads() around dependent calls.
// ---------------------------------------------------------------------------
__device__ __forceinline__ void wg_gemm64(const float* A, const float* B,
                                          float* D, bool transB, int tid) {
  const int lane = tid & 31;
  const int wave = tid >> 5;
  const int hl   = lane >> 4;   // half-wave select
  const int l16  = lane & 15;
  const int tn   = (wave & 3) << 4;
  const int tm0  = (wave >> 2) << 5;
  const int tm1  = tm0 + 16;
  v8f c0 = {0.f, 0.f, 0.f, 0.f, 0.f, 0.f, 0.f, 0.f};
  v8f c1 = {0.f, 0.f, 0.f, 0.f, 0.f, 0.f, 0.f, 0.f};
#if defined(__HIP_DEVICE_COMPILE__)
#pragma unroll
  for (int k0 = 0; k0 < 64; k0 += 4) {
    const int ak = k0 + 2 * hl;
    const int bn = tn + l16;
    v2f b;
    if (transB) {
      b = *(const v2f*)&B[bn * LDSS + ak];       // B^T: contiguous in k (8B aligned)
    } else {
      b.x = B[(ak + 0) * LDSS + bn];
      b.y = B[(ak + 1) * LDSS + bn];
    }
    v2f a0 = *(const v2f*)&A[(tm0 + l16) * LDSS + ak];
    v2f a1 = *(const v2f*)&A[(tm1 + l16) * LDSS + ak];
    c0 = __builtin_amdgcn_wmma_f32_16x16x4_f32(false, a0, false, b,
                                               (short)0, c0, false, false);
    c1 = __builtin_amdgcn_wmma_f32_16x16x4_f32(false, a1, false, b,
                                               (short)0, c1, false, false);
  }
#endif
#pragma unroll
  for (int r = 0; r < 8; ++r) {
    D[(tm0 + r + 8 * hl) * LDSS + tn + l16] = c0[r];
    D[(tm1 + r + 8 * hl) * LDSS + tn + l16] = c1[r];
  }
}

// Z = a*X + b*Y + d*I  (elementwise over 64x64; Y may be nullptr; Z==X ok)
__device__ __forceinline__ void lds_axpby(float* Z, float a, const float* X,
                                          float b, const float* Y, float d, int tid) {
  for (int i = tid; i < 4096; i += WG) {
    const int r = i >> 6, c = i & 63, o = r * LDSS + c;
    float v = a * X[o];
    if (Y) v += b * Y[o];
    if (r == c) v += d;
    Z[o] = v;
  }
}

__device__ __forceinline__ void lds_seti(float* Z, float d, int tid) {
  for (int i = tid; i < 4096; i += WG) {
    const int r = i >> 6, c = i & 63;
    Z[r * LDSS + c] = (r == c) ? d : 0.f;
  }
}

// Load rows x cols (leading dim ld) into padded 64x64 LDS tile; pad diag = diag.
__device__ __forceinline__ void lds_ld_global(float* dst, const float* __restrict__ src,
                                              int rows, int cols, int ld, float diag, int tid) {
  for (int i = tid; i < 4096; i += WG) {
    const int r = i >> 6, c = i & 63;
    float v = (r < rows && c < cols) ? src[r * ld + c] : 0.f;
    if (r == c && r >= rows) v = diag;
    dst[r * LDSS + c] = v;
  }
}

// Store 64x64 LDS tile; add diagAdd on diag entries at/after liveDim.
__device__ __forceinline__ void lds_st_global(float* __restrict__ dst, const float* src,
                                              float diagAdd, int liveDim, int tid) {
  for (int i = tid; i < 4096; i += WG) {
    const int r = i >> 6, c = i & 63;
    float v = src[r * LDSS + c];
    if (r == c && r >= liveDim) v += diagAdd;
    dst[i] = v;
  }
}

// Coupled Newton-Schulz sqrt: entry y=C (SPD, pad=I). Exit y=sqrt(C), z=isqrt(C).
// t,u scratch; pointers rotate (by reference). Trace pre-normalization.
__device__ __forceinline__ void ns_sqrt_pair(float*& y, float*& z, float*& t, float*& u,
                                             float* red, int tid, int iters) {
  if (tid == 0) {
    float s = 0.f;
    for (int k = 0; k < 64; ++k) s += y[k * LDSS + k];
    red[0] = s;
  }
  __syncthreads();
  const float c = red[0];
  lds_axpby(y, 1.f / c, y, 0.f, nullptr, 0.f, tid);   // y = C / tr(C)
  lds_seti(z, 1.f, tid);
  __syncthreads();
  for (int it = 0; it < iters; ++it) {
    wg_gemm64(z, y, t, false, tid); __syncthreads();
    lds_axpby(t, -0.5f, t, 0.f, nullptr, 1.5f, tid); __syncthreads();  // t = (3I - ZY)/2
    wg_gemm64(y, t, u, false, tid); __syncthreads();                    // newY
    wg_gemm64(t, z, y, false, tid); __syncthreads();                    // newZ (into old y buf)
    float* oy = y; float* oz = z;
    y = u; z = oy; u = oz;                                              // rotate
  }
  const float sc = sqrtf(c);
  lds_axpby(y, sc, y, 0.f, nullptr, 0.f, tid);
  lds_axpby(z, 1.f / sc, z, 0.f, nullptr, 0.f, tid);
  __syncthreads();
}

// logm via inverse scaling-and-squaring: 3 NS sqrts + Horner log(I+E), x8.
// Entry: y = C. Returns pointer holding L/8 (caller multiplies by 8).
__device__ __forceinline__ float* logm64(float*& y, float*& z, float*& t, float*& u,
                                         float* red, int tid) {
  for (int sq = 0; sq < 3; ++sq) ns_sqrt_pair(y, z, t, u, red, tid, 9);
  lds_axpby(y, 1.f, y, 0.f, nullptr, -1.f, tid); __syncthreads();   // E = C^(1/8) - I
  float* E = y; float* H = z; float* Wb = u;
  lds_axpby(H, -1.f / 6.f, E, 0.f, nullptr, 0.2f, tid); __syncthreads();  // H = I/5 - E/6
  const float kc[3] = {-0.25f, 1.f / 3.f, -0.5f};
  for (int j = 0; j < 3; ++j) {
    wg_gemm64(E, H, Wb, false, tid); __syncthreads();
    lds_axpby(Wb, 1.f, Wb, 0.f, nullptr, kc[j], tid); __syncthreads();
    float* tm = H; H = Wb; Wb = tm;
  }
  wg_gemm64(E, H, Wb, false, tid); __syncthreads();
  lds_axpby(Wb, 1.f, Wb, 0.f, nullptr, 1.f, tid); __syncthreads();  // S = I + E*H
  wg_gemm64(E, Wb, t, false, tid); __syncthreads();                 // L/8 = E*S
  return t;
}

// ---------------------------------------------------------------------------
// Kernels
// ---------------------------------------------------------------------------

// Y = W * X * W^T per sample; output padded with +padDiag beyond liveOut.
__global__ __launch_bounds__(WG) void congruence_kernel(
    const float* __restrict__ X, int inDim, int inLd, size_t inStride,
    const float* __restrict__ W, float* __restrict__ out,
    int liveOut, float padDiag) {
  extern __shared__ float sm[];
  float* Wl = sm; float* Xl = sm + MS; float* Tl = sm + 2 * MS;
  const int s = blockIdx.x, tid = threadIdx.x;
  lds_ld_global(Wl, W, 64, 64, 64, 0.f, tid);
  lds_ld_global(Xl, X + (size_t)s * inStride, inDim, inDim, inLd, 0.f, tid);
  __syncthreads();
  wg_gemm64(Wl, Xl, Tl, false, tid); __syncthreads();   // T = W X
  wg_gemm64(Tl, Wl, Xl, true, tid);  __syncthreads();   // Y = T W^T
  lds_st_global(out + (size_t)s * NMAT, Xl, padDiag, liveOut, tid);
}

__global__ void mean_kernel(const float* __restrict__ buf, float* __restrict__ out, int N) {
  const int j = blockIdx.x * blockDim.x + threadIdx.x;
  if (j >= NMAT) return;
  float s = 0.f;
  for (int n = 0; n < N; ++n) s += buf[(size_t)n * NMAT + j];
  out[j] = s / (float)N;   // mean of SPD batch is symmetric already
}

__global__ void zero_kernel(float* __restrict__ p, int n) {
  const int i = blockIdx.x * blockDim.x + threadIdx.x;
  if (i < n) p[i] = 0.f;
}

__global__ void pad_kernel(const float* __restrict__ src, int rows, int cols,
                           float diag, float* __restrict__ dst) {
  const int i = blockIdx.x * blockDim.x + threadIdx.x;
  if (i >= NMAT) return;
  const int r = i >> 6, c = i & 63;
  float v = (r < rows && c < cols) ? src[r * cols + c] : 0.f;
  if (r == c && r >= rows) v = diag;
  dst[i] = v;
}

// Per-sample: accum += (1/N) * logm(P * X * P)   (P symmetric, e.g. M0^-1/2)
__global__ __launch_bounds__(WG) void logm_accum_kernel(
    const float* __restrict__ X, const float* __restrict__ P,
    float* __restrict__ accum, float mul) {
  extern __shared__ float sm[];
  float* b0 = sm; float* b1 = sm + MS; float* b2 = sm + 2 * MS; float* b3 = sm + 3 * MS;
  float* red = sm + 4 * MS;
  const int s = blockIdx.x, tid = threadIdx.x;
  lds_ld_global(b0, X + (size_t)s * NMAT, 64, 64, 64, 0.f, tid);
  lds_ld_global(b1, P, 64, 64, 64, 0.f, tid);
  __syncthreads();
  wg_gemm64(b1, b0, b2, false, tid); __syncthreads();   // P X
  wg_gemm64(b2, b1, b3, false, tid); __syncthreads();   // C = P X P
  float* y = b3; float* z = b0; float* t = b1; float* u = b2;
  float* L = logm64(y, z, t, u, red, tid);              // L/8
  for (int i = tid; i < 4096; i += WG) {
    const int r = i >> 6, c = i & 63;
    atomicAdd(accum + i, L[r * LDSS + c] * mul);        // mul = 8/N
  }
}

// Per-sample ReEig via matrix sign: 0.5*((X-eI)sign(X-eI) + X + eI)
__global__ __launch_bounds__(WG) void reeig_kernel(
    const float* __restrict__ X, float* __restrict__ out, float eps) {
  extern __shared__ float sm[];
  float* bX = sm; float* bB = sm + MS; float* bZ = sm + 2 * MS;
  float* bT = sm + 3 * MS; float* bU = sm + 4 * MS;
  float* red = sm + 5 * MS;
  const int s = blockIdx.x, tid = threadIdx.x;
  lds_ld_global(bX, X + (size_t)s * NMAT, 64, 64, 64, 0.f, tid);
  __syncthreads();
  lds_axpby(bB, 1.f, bX, 0.f, nullptr, -eps, tid);      // B = X - eps I
  if (tid == 0) red[0] = 0.f;
  __syncthreads();
  float p = 0.f;
  for (int i = tid; i < 4096; i += WG) {
    const int r = i >> 6, c = i & 63;
    const float v = bB[r * LDSS + c];
    p += v * v;
  }
  atomicAdd(red, p);
  __syncthreads();
  const float f = sqrtf(red[0]) + 1e-30f;
  lds_axpby(bZ, 1.f / f, bB, 0.f, nullptr, 0.f, tid);   // Z0 = B/||B||_F
  __syncthreads();
  float* Z = bZ; float* U = bU;
  for (int it = 0; it < 14; ++it) {                     // Z <- Z(3I - Z^2)/2
    wg_gemm64(Z, Z, bT, false, tid); __syncthreads();
    lds_axpby(bT, -0.5f, bT, 0.f, nullptr, 1.5f, tid); __syncthreads();
    wg_gemm64(Z, bT, U, false, tid); __syncthreads();
    float* tm = Z; Z = U; U = tm;
  }
  wg_gemm64(bB, Z, bT, false, tid); __syncthreads();    // |B| = B*sign(B)
  for (int i = tid; i < 4096; i += WG) {
    const int r = i >> 6, c = i & 63;
    float v = 0.5f * (bT[r * LDSS + c] + bX[r * LDSS + c]);
    if (r == c) v += 0.5f * eps;
    out[(size_t)s * NMAT + i] = v;
  }
}

// Per-sample LogEig + linear head: y = Wl @ vec(logm(feat)[0:50,0:50]) + bl
__global__ __launch_bounds__(WG) void logm_final_kernel(
    const float* __restrict__ X, const float* __restrict__ Wl,
    const float* __restrict__ bl, float* __restrict__ yOut) {
  extern __shared__ float sm[];
  float* b0 = sm; float* b1 = sm + MS; float* b2 = sm + 2 * MS; float* b3 = sm + 3 * MS;
  float* red = sm + 4 * MS;
  const int s = blockIdx.x, tid = threadIdx.x;
  lds_ld_global(b0, X + (size_t)s * NMAT, 64, 64, 64, 0.f, tid);
  __syncthreads();
  float* y = b0; float* z = b1; float* t = b2; float* u = b3;
  float* L = logm64(y, z, t, u, red, tid);              // L/8
  if (tid == 0) { red[1] = 0.f; red[2] = 0.f; }
  __syncthreads();
  float p0 = 0.f, p1 = 0.f;
  for (int i = tid; i < 4096; i += WG) {
    const int r = i >> 6, c = i & 63;
    if (r < 50 && c < 50) {
      const float v = 8.f * L[r * LDSS + c];
      const int k = r * 50 + c;
      p0 += Wl[k] * v;
      p1 += Wl[2500 + k] * v;
    }
  }
  atomicAdd(&red[1], p0);
  atomicAdd(&red[2], p1);
  __syncthreads();
  if (tid == 0) {
    yOut[s * 2 + 0] = red[1] + bl[0];
    yOut[s * 2 + 1] = red[2] + bl[1];
  }
}

// Extract 50x50 live block of feat into output layout (N,1,50,50)
__global__ void feat_copy_kernel(const float* __restrict__ feat, float* __restrict__ out, int total) {
  const int i = blockIdx.x * blockDim.x + threadIdx.x;
  if (i >= total) return;
  const int s = i / 2500, rem = i % 2500, r = rem / 50, c = rem % 50;
  out[i] = feat[(size_t)s * NMAT + r * 64 + c];
}

// Single-workgroup matrix-function kernel on 64x64 matrices:
//  op0: out0 = sqrtm(in0), out1 = isqrtm(in0)
//  op1: out0 = sym(in1 * expm(in0) * in1)          (in0=L, in1=M0^1/2)
//  op2: out0 = in0 * in1
__global__ __launch_bounds__(WG) void matfun_kernel(
    int op, const float* __restrict__ in0, const float* __restrict__ in1,
    float* __restrict__ out0, float* __restrict__ out1) {
  extern __shared__ float sm[];
  float* b0 = sm;          float* b1 = sm + MS;     float* b2 = sm + 2 * MS;
  float* b3 = sm + 3 * MS; float* b4 = sm + 4 * MS; float* b5 = sm + 5 * MS;
  float* red = sm + 6 * MS;
  const int tid = threadIdx.x;
  if (op == 0) {
    lds_ld_global(b0, in0, 64, 64, 64, 0.f, tid);
    __syncthreads();
    float* y = b0; float* z = b1; float* t = b2; float* u = b3;
    ns_sqrt_pair(y, z, t, u, red, tid, 14);
    lds_st_global(out0, y, 0.f, 64, tid);
    if (out1) lds_st_global(out1, z, 0.f, 64, tid);
  } else if (op == 1) {
    lds_ld_global(b0, in0, 64, 64, 64, 0.f, tid);   // L
    lds_ld_global(b1, in1, 64, 64, 64, 0.f, tid);   // M = M0^1/2
    __syncthreads();
    lds_axpby(b0, 1.f / 16.f, b0, 0.f, nullptr, 0.f, tid); __syncthreads(); // A = L/16
    lds_axpby(b2, 1.f, b0, 0.f, nullptr, 1.f, tid);                          // E = I + A
    lds_axpby(b3, 1.f, b0, 0.f, nullptr, 0.f, tid); __syncthreads();         // Pw = A
    for (int k = 2; k <= 6; ++k) {                   // Taylor of exp(A)
      wg_gemm64(b3, b0, b4, false, tid); __syncthreads();
      lds_axpby(b3, 1.f / (float)k, b4, 0.f, nullptr, 0.f, tid); __syncthreads();
      lds_axpby(b2, 1.f, b2, 1.f, b3, 0.f, tid); __syncthreads();
    }
    for (int sq = 0; sq < 4; ++sq) {                 // square back: (e^{A})^{16}
      wg_gemm64(b2, b2, b4, false, tid); __syncthreads();
      lds_axpby(b2, 1.f, b4, 0.f, nullptr, 0.f, tid); __syncthreads();
    }
    wg_gemm64(b1, b2, b4, false, tid); __syncthreads();   // M * expm(L)
    wg_gemm64(b4, b1, b5, false, tid); __syncthreads();   // G' = M expm(L) M
    for (int i = tid; i < 4096; i += WG) {
      const int r = i >> 6, c = i & 63;
      out0[i] = 0.5f * (b5[r * LDSS + c] + b5[c * LDSS + r]);  // sym
    }
  } else {  // op2
    lds_ld_global(b0, in0, 64, 64, 64, 0.f, tid);
    lds_ld_global(b1, in1, 64, 64, 64, 0.f, tid);
    __syncthreads();
    wg_gemm64(b0, b1, b2, false, tid); __syncthreads();
    lds_st_global(out0, b2, 0.f, 64, tid);
  }
}

// ---------------------------------------------------------------------------
// Host orchestration
// ---------------------------------------------------------------------------
extern "C" void kernel_launch(void* const* d_in, const int* in_sizes, int n_in,
                              void* d_out, int out_size, void* d_ws, size_t ws_size,
                              hipStream_t stream) {
  const float* x  = (const float*)d_in[0];   // (8192,1,62,62)
  const float* W1 = (const float*)d_in[1];   // (58,62)
  const float* W2 = (const float*)d_in[2];   // (54,58)
  const float* W3 = (const float*)d_in[3];   // (50,54)
  const float* G1 = (const float*)d_in[4];   // (58,58)
  const float* G2 = (const float*)d_in[5];   // (54,54)
  const float* G3 = (const float*)d_in[6];   // (50,50)
  const float* Wl = (const float*)d_in[7];   // (2,2500)
  const float* bl = (const float*)d_in[8];   // (2,)
  float* out = (float*)d_out;                // [y (8192*2)] ++ [feat (8192*2500)]
  (void)in_sizes; (void)n_in; (void)out_size; (void)ws_size;

  const int N = NSAMP;
  float* ws   = (float*)d_ws;
  float* bufA = ws;                                  // N * 4096
  float* bufB = ws + (size_t)N * NMAT;               // N * 4096
  float* smat = ws + 2 * (size_t)N * NMAT;           // small 64x64 slots
  auto slot = [&](int i) { return smat + (size_t)i * NMAT; };
  // slots: 0 W1p 1 W2p 2 W3p 3 G1p 4 G2p 5 G3p 6 M0 7 M0s 8 M0is 9 Lsum
  //        10 G 11 Gs 12 Gis 13 Ws 14 Wis 15 P

  const size_t SH3 = (size_t)3 * MS * sizeof(float);
  const size_t SH4 = ((size_t)4 * MS + 8) * sizeof(float);
  const size_t SH5 = ((size_t)5 * MS + 8) * sizeof(float);
  const size_t SH6 = ((size_t)6 * MS + 8) * sizeof(float);
  const float invN = 1.f / (float)N;

  // Pad weights into 64x64 (rect BiMap weights: zero pad; SPD biases: identity pad)
  pad_kernel<<<16, 256, 0, stream>>>(W1, 58, 62, 0.f, slot(0));
  pad_kernel<<<16, 256, 0, stream>>>(W2, 54, 58, 0.f, slot(1));
  pad_kernel<<<16, 256, 0, stream>>>(W3, 50, 54, 0.f, slot(2));
  pad_kernel<<<16, 256, 0, stream>>>(G1, 58, 58, 1.f, slot(3));
  pad_kernel<<<16, 256, 0, stream>>>(G2, 54, 54, 1.f, slot(4));
  pad_kernel<<<16, 256, 0, stream>>>(G3, 50, 50, 1.f, slot(5));

  // Riemannian batch-norm: src -> dst  (dst = Ws Gis src Gis Ws = P src P^T)
  auto bn = [&](const float* src, float* dst, const float* Gp) {
    mean_kernel<<<16, 256, 0, stream>>>(src, slot(6), N);                       // M0
    matfun_kernel<<<1, 256, SH6, stream>>>(0, slot(6), nullptr, slot(7), slot(8)); // M0s,M0is
    zero_kernel<<<16, 256, 0, stream>>>(slot(9), NMAT);
    logm_accum_kernel<<<N, 256, SH4, stream>>>(src, slot(8), slot(9), 8.f * invN); // L
    matfun_kernel<<<1, 256, SH6, stream>>>(1, slot(9), slot(7), slot(10), nullptr); // G
    matfun_kernel<<<1, 256, SH6, stream>>>(0, slot(10), nullptr, slot(11), slot(12)); // Gis
    matfun_kernel<<<1, 256, SH6, stream>>>(0, Gp, nullptr, slot(13), slot(14));       // Ws
    matfun_kernel<<<1, 256, SH6, stream>>>(2, slot(13), slot(12), slot(15), nullptr); // P=Ws*Gis
    congruence_kernel<<<N, 256, SH3, stream>>>(src, 64, 64, (size_t)NMAT,
                                               slot(15), dst, 64, 0.f);
  };

  // Stage 1: bimap(62->58) -> BN -> ReEig
  congruence_kernel<<<N, 256, SH3, stream>>>(x, 62, 62, (size_t)62 * 62,
                                             slot(0), bufA, 58, 1.f);
  bn(bufA, bufB, slot(3));
  reeig_kernel<<<N, 256, SH5, stream>>>(bufB, bufA, 1e-4f);

  // Stage 2: bimap(58->54) -> BN -> ReEig
  congruence_kernel<<<N, 256, SH3, stream>>>(bufA, 64, 64, (size_t)NMAT,
                                             slot(1), bufB, 54, 1.f);
  bn(bufB, bufA, slot(4));
  reeig_kernel<<<N, 256, SH5, stream>>>(bufA, bufB, 1e-4f);

  // Stage 3: bimap(54->50) -> BN (no ReEig); feat in bufB
  congruence_kernel<<<N, 256, SH3, stream>>>(bufB, 64, 64, (size_t)NMAT,
                                             slot(2), bufA, 50, 1.f);
  bn(bufA, bufB, slot(5));

  // Outputs: y (logits) then feat
  const int featTot = N * 2500;
  feat_copy_kernel<<<(featTot + 255) / 256, 256, 0, stream>>>(bufB, out + 2 * N, featTot);
  logm_final_kernel<<<N, 256, SH4, stream>>>(bufB, Wl, bl, out);
}